// FusedTensorProductOp3_52879637348700
// MI455X (gfx1250) — compile-verified
//
#include <hip/hip_runtime.h>

#define NSEG   64
#define SEGSZ  64
#define NPATH  256
#define FEAT   (NSEG * SEGSZ)   // 4096
#define ZPB    2                // batch rows per block
#define TPB    64               // 2 waves (wave32); 32 channel-pair lanes per z

typedef float  v2f __attribute__((ext_vector_type(2)));

// Truncate a generic pointer to an LDS byte offset (ISA 10.2: LDS aperture
// uses addr[31:0] as the wave-relative LDS address).
__device__ __forceinline__ unsigned lds_offset(const void* p) {
    return (unsigned)(unsigned long long)p;
}

__device__ __forceinline__ void async_load_b128_to_lds(unsigned lds_off, const void* gaddr) {
    asm volatile("global_load_async_to_lds_b128 %0, %1, off"
                 :: "v"(lds_off), "v"(gaddr)
                 : "memory");
}

__device__ __forceinline__ void wait_asynccnt0() {
    asm volatile("s_wait_asynccnt 0" ::: "memory");
}

// Packed per-path record: offsets in float2 units (seg*32 each, <= 2016
// fits in 16 bits) + coefficient. One ds_load_b64 broadcast per path.
struct __align__(8) PathMeta {
    unsigned off;   // [15:0] = i0*32, [31:16] = i1*32
    float    c;
};

__global__ __launch_bounds__(TPB) void seg_tp_kernel(
    const float* __restrict__ x0,
    const float* __restrict__ x1,
    const int*   __restrict__ pi0,
    const int*   __restrict__ pi1,
    const int*   __restrict__ pi2,
    const float* __restrict__ pc,
    float*       __restrict__ out)
{
    // x tiles for ZPB batch rows: 2 * 2 * 4096 * 4B = 64 KB
    __shared__ __align__(16) float xs0[ZPB][FEAT];
    __shared__ __align__(16) float xs1[ZPB][FEAT];
    // path metadata, bucketed by output segment
    __shared__ int      perm[NPATH];
    __shared__ int      segstart[NSEG + 1];
    __shared__ PathMeta meta[NPATH];
    __shared__ int      cnt[NSEG];

    const int tid = threadIdx.x;
    const int z0  = blockIdx.x * ZPB;

    // ------------------------------------------------------------------
    // 1) Async global->LDS staging of the two x tiles.
    //    ZPB*FEAT = 8192 floats = 2048 b128 chunks per tensor.
    //    64 threads -> 32 chunks per tensor per thread (hardware stalls
    //    issue if ASYNCcnt would overflow, so >63 in flight is safe).
    // ------------------------------------------------------------------
    {
        const float* g0 = x0 + (size_t)z0 * FEAT;
        const float* g1 = x1 + (size_t)z0 * FEAT;
        const unsigned l0 = lds_offset(&xs0[0][0]);
        const unsigned l1 = lds_offset(&xs1[0][0]);
        for (int i = tid; i < (ZPB * FEAT) / 4; i += TPB) {
            const unsigned boff = (unsigned)i * 16u;
            async_load_b128_to_lds(l0 + boff, g0 + i * 4);
            async_load_b128_to_lds(l1 + boff, g1 + i * 4);
        }
    }

    // ------------------------------------------------------------------
    // 2) Overlap: counting-sort the 256 paths by destination segment i2.
    //    Turns the scatter-add into a dense per-segment register reduction.
    // ------------------------------------------------------------------
    if (tid < NSEG) {
        int c = 0;
        for (int p = 0; p < NPATH; ++p) c += (pi2[p] == tid) ? 1 : 0;
        cnt[tid] = c;
    }
    __syncthreads();
    if (tid == 0) {
        int acc = 0;
        for (int s = 0; s < NSEG; ++s) { segstart[s] = acc; acc += cnt[s]; }
        segstart[NSEG] = acc;   // == NPATH
    }
    __syncthreads();
    if (tid < NSEG) {
        int w = segstart[tid];
        for (int p = 0; p < NPATH; ++p) {
            if (pi2[p] == tid) perm[w++] = p;
        }
    }
    __syncthreads();
    for (int k = tid; k < NPATH; k += TPB) {
        const int p = perm[k];
        const unsigned o0 = (unsigned)(pi0[p] * (SEGSZ / 2));   // float2 units
        const unsigned o1 = (unsigned)(pi1[p] * (SEGSZ / 2));
        meta[k].off = o0 | (o1 << 16);
        meta[k].c   = pc[p];
    }

    // Wait for this wave's async tile loads, then barrier so every wave
    // sees both the x tiles and the bucketed path metadata.
    wait_asynccnt0();
    __syncthreads();

    // ------------------------------------------------------------------
    // 3) Compute: thread owns (z-local, channel pair). For each output
    //    segment, reduce its paths in a float2 register, store once.
    //    ds_load_b64 gathers touch all 64 banks exactly once per wave.
    // ------------------------------------------------------------------
    const int zl = tid >> 5;          // 0..ZPB-1
    const int uh = tid & 31;          // channel pair index 0..31
    const v2f* __restrict__ a2 = reinterpret_cast<const v2f*>(&xs0[zl][0]);
    const v2f* __restrict__ b2 = reinterpret_cast<const v2f*>(&xs1[zl][0]);
    v2f* __restrict__ orow =
        reinterpret_cast<v2f*>(out + (size_t)(z0 + zl) * FEAT);

    for (int s = 0; s < NSEG; ++s) {
        const int k0 = segstart[s];
        const int k1 = segstart[s + 1];
        v2f acc = {0.0f, 0.0f};
        for (int k = k0; k < k1; ++k) {
            const unsigned off = meta[k].off;
            const float    c   = meta[k].c;
            const v2f av = a2[(off & 0xFFFFu) + uh];
            const v2f bv = b2[(off >> 16) + uh];
            acc.x = fmaf(c, av.x * bv.x, acc.x);
            acc.y = fmaf(c, av.y * bv.y, acc.y);
        }
        // Write-once output, never re-read by this kernel: bypass caches.
        __builtin_nontemporal_store(acc, &orow[s * (SEGSZ / 2) + uh]);
    }
}

extern "C" void kernel_launch(void* const* d_in, const int* in_sizes, int n_in,
                              void* d_out, int out_size, void* d_ws, size_t ws_size,
                              hipStream_t stream) {
    (void)n_in; (void)out_size; (void)d_ws; (void)ws_size;
    const float* x0  = (const float*)d_in[0];
    const float* x1  = (const float*)d_in[1];
    const int*   pi0 = (const int*)d_in[2];
    const int*   pi1 = (const int*)d_in[3];
    const int*   pi2 = (const int*)d_in[4];
    const float* pc  = (const float*)d_in[5];
    float* out = (float*)d_out;

    const int Z = in_sizes[0] / FEAT;     // 4096
    const int grid = Z / ZPB;             // 2048 blocks
    seg_tp_kernel<<<grid, TPB, 0, stream>>>(x0, x1, pi0, pi1, pi2, pc, out);
}